// GraphBackbone_72997264163107
// MI455X (gfx1250) — compile-verified
//
#include <hip/hip_runtime.h>
#include <math.h>

#define N_NODES 50000
#define N_EDGES 800000
#define DHID 64

typedef __attribute__((ext_vector_type(2))) float v2f;
typedef __attribute__((ext_vector_type(8))) float v8f;

// ---- monotonic float<->uint encoding for atomicMax on f32 ----
__device__ __forceinline__ unsigned fenc(float f) {
    unsigned u = __float_as_uint(f);
    return (u >> 31) ? ~u : (u | 0x80000000u);
}
__device__ __forceinline__ float fdec(unsigned u) {
    return __uint_as_float((u & 0x80000000u) ? (u & 0x7FFFFFFFu) : ~u);
}
#define FENC_NEG_INF 0x007FFFFFu   // fenc(-INFINITY)

// =====================================================================
// One-time prep: pad node features (N x 7) -> (N x 8, zero padded)
// =====================================================================
__global__ __launch_bounds__(256)
void pad_x(const float* __restrict__ xin, float* __restrict__ xpad, int N)
{
    const int i = blockIdx.x * blockDim.x + threadIdx.x;
    if (i >= N * 8) return;
    const int n = i >> 3, c = i & 7;
    xpad[i] = (c < 7) ? xin[n * 7 + c] : 0.0f;
}

// =====================================================================
// Per-layer prep: pack [Wq|Wk|Wv|Wskip] -> Wcat (KP x 256, zero-padded
// rows K..KP-1) and biases -> bcat (256). Tiny one-shot kernel.
// =====================================================================
__global__ __launch_bounds__(256)
void pack_weights(const float* __restrict__ Wq, const float* __restrict__ Wk,
                  const float* __restrict__ Wv, const float* __restrict__ Ws,
                  const float* __restrict__ bq, const float* __restrict__ bk,
                  const float* __restrict__ bv, const float* __restrict__ bs,
                  float* __restrict__ Wcat, float* __restrict__ bcat,
                  int K, int KP)
{
    const int i = blockIdx.x * blockDim.x + threadIdx.x;
    if (i < KP * 256) {
        const int k = i >> 8, c = i & 255;
        const int mat = c >> 6, cc = c & 63;
        const float* W = (mat == 0) ? Wq : (mat == 1) ? Wk : (mat == 2) ? Wv : Ws;
        Wcat[i] = (k < K) ? W[k * DHID + cc] : 0.0f;
    }
    if (i < 256) {
        const int mat = i >> 6, cc = i & 63;
        const float* B = (mat == 0) ? bq : (mat == 1) ? bk : (mat == 2) ? bv : bs;
        bcat[i] = B[cc];
    }
}

// =====================================================================
// Fused node GEMM: O[4][N][64] = X(N x KP) @ Wcat(KP x 256) + bcat.
// f32 WMMA 16x16x4, wave32. Block = 256 thr = 8 waves; block covers a
// 16-row x 256-col output strip; each wave owns two 16x16 tiles.
// A tile staged in LDS (coalesced b32 in, ds_load_b64 fragments out).
// No guards: N % 16 == 0, KP % 4 == 0, rows K..KP-1 of Wcat are zero.
// =====================================================================
template<int KP>
__global__ __launch_bounds__(256)
void gemm_qkvs(const float* __restrict__ X, const float* __restrict__ Wcat,
               const float* __restrict__ bcat, float* __restrict__ O, int Nn)
{
    __shared__ float sA[16 * KP];           // <= 4 KB
    const int m0 = blockIdx.x * 16;

    for (int i = threadIdx.x; i < 16 * KP; i += 256)
        sA[i] = X[(size_t)m0 * KP + i];     // rows are contiguous
    __syncthreads();

    const int lane  = threadIdx.x & 31;
    const int wave  = threadIdx.x >> 5;
    const int lm    = lane & 15;            // M (A) / N (B, D)
    const int lhalf = lane >> 4;            // K-group select / M+8 for D

    #pragma unroll
    for (int tt = 0; tt < 2; ++tt) {
        const int t   = 2 * wave + tt;      // col tile 0..15
        const int mat = t >> 2;             // Q/K/V/S
        const int nc  = (t & 3) * 16 + lm;  // col within 64
        const int ncg = t * 16 + lm;        // col within 256

        v8f c = {};
        #pragma unroll
        for (int k = 0; k < KP; k += 4) {
            const int ka = k + lhalf * 2;
            v2f a, b;
            a.x = sA[lm * KP + ka];
            a.y = sA[lm * KP + ka + 1];
            b.x = Wcat[ka * 256 + ncg];
            b.y = Wcat[(ka + 1) * 256 + ncg];
            c = __builtin_amdgcn_wmma_f32_16x16x4_f32(
                    false, a, false, b, (short)0, c, false, false);
        }
        const float bb = bcat[ncg];
        float* Om = O + (size_t)mat * (size_t)Nn * DHID;
        #pragma unroll
        for (int j = 0; j < 8; ++j)
            Om[(size_t)(m0 + j + lhalf * 8) * DHID + nc] = c[j] + bb;
    }
}

// =====================================================================
// Init per-layer accumulators: out=0, denom=0, segmax=enc(-inf)
// =====================================================================
__global__ __launch_bounds__(256)
void seg_init(float* __restrict__ out, float* __restrict__ denom,
              unsigned* __restrict__ segmax, int n_out, int n_seg)
{
    const int i = blockIdx.x * blockDim.x + threadIdx.x;
    if (i < n_out) out[i] = 0.0f;
    if (i < n_seg) { denom[i] = 0.0f; segmax[i] = FENC_NEG_INF; }
}

// =====================================================================
// Pass A: alpha[e,h] = dot(q[dst],k[src]+ea*We)/sqrt(C) + lam*cpn[dst]
// =====================================================================
template<int H, int C>
__global__ __launch_bounds__(256)
void edge_alpha(const int* __restrict__ src, const int* __restrict__ dst,
                const float* __restrict__ eattr, const float* __restrict__ cpn,
                const float* __restrict__ We, const float* __restrict__ lam,
                const float* __restrict__ Q, const float* __restrict__ Kn,
                float* __restrict__ alpha, unsigned* __restrict__ segmax, int E)
{
    __shared__ float sWe[DHID];
    if (threadIdx.x < DHID) sWe[threadIdx.x] = We[threadIdx.x];
    __syncthreads();

    const int e = blockIdx.x * blockDim.x + threadIdx.x;
    if (e >= E) return;
    const int s = src[e], d = dst[e];
    const float ea = log1pf(eattr[e]);
    const float cb = lam[0] * cpn[d];
    const float rs = 1.0f / sqrtf((float)C);

    #pragma unroll
    for (int h = 0; h < H; ++h) {
        float acc = 0.0f;
        #pragma unroll
        for (int c = 0; c < C; c += 4) {
            const float4 qv = *(const float4*)&Q [(size_t)d * DHID + h * C + c];
            const float4 kv = *(const float4*)&Kn[(size_t)s * DHID + h * C + c];
            const float4 wv = *(const float4*)&sWe[h * C + c];
            acc += qv.x * (kv.x + ea * wv.x) + qv.y * (kv.y + ea * wv.y)
                 + qv.z * (kv.z + ea * wv.z) + qv.w * (kv.w + ea * wv.w);
        }
        const float a = acc * rs + cb;
        alpha[(size_t)e * H + h] = a;
        atomicMax(&segmax[(size_t)d * H + h], fenc(a));
    }
}

// =====================================================================
// Pass B: ex = exp(alpha - max[dst]); alpha <- ex; denom += ex
// =====================================================================
template<int H>
__global__ __launch_bounds__(256)
void edge_exp(const int* __restrict__ dst, float* __restrict__ alpha,
              const unsigned* __restrict__ segmax, float* __restrict__ denom, int E)
{
    const int i = blockIdx.x * blockDim.x + threadIdx.x;
    if (i >= E * H) return;
    const int e = i / H, h = i - e * H;
    const int d = dst[e];
    const float m  = fdec(segmax[(size_t)d * H + h]);
    const float ex = expf(alpha[i] - m);
    alpha[i] = ex;
    atomicAdd(&denom[(size_t)d * H + h], ex);
}

// =====================================================================
// Pass C: out[dst] += (ex/denom) * (v[src] + ea*We)
// =====================================================================
template<int H, int C>
__global__ __launch_bounds__(256)
void edge_scatter(const int* __restrict__ src, const int* __restrict__ dst,
                  const float* __restrict__ eattr, const float* __restrict__ We,
                  const float* __restrict__ alpha, const float* __restrict__ denom,
                  const float* __restrict__ V, float* __restrict__ out, int E)
{
    __shared__ float sWe[DHID];
    if (threadIdx.x < DHID) sWe[threadIdx.x] = We[threadIdx.x];
    __syncthreads();

    const int e = blockIdx.x * blockDim.x + threadIdx.x;
    if (e >= E) return;
    const int s = src[e], d = dst[e];
    const float ea = log1pf(eattr[e]);

    #pragma unroll
    for (int h = 0; h < H; ++h) {
        const float w = alpha[(size_t)e * H + h]
                      / (denom[(size_t)d * H + h] + 1e-16f);
        #pragma unroll
        for (int c = 0; c < C; c += 4) {
            const float4 vv = *(const float4*)&V[(size_t)s * DHID + h * C + c];
            const float4 wv = *(const float4*)&sWe[h * C + c];
            float* o = &out[(size_t)d * DHID + h * C + c];
            atomicAdd(o + 0, w * (vv.x + ea * wv.x));
            atomicAdd(o + 1, w * (vv.y + ea * wv.y));
            atomicAdd(o + 2, w * (vv.z + ea * wv.z));
            atomicAdd(o + 3, w * (vv.w + ea * wv.w));
        }
    }
}

// =====================================================================
// beta = sigmoid([out, x_r, out-x_r] @ Wbeta); x = beta*x_r+(1-beta)*out
// =====================================================================
__global__ __launch_bounds__(256)
void node_combine(const float* __restrict__ out, const float* __restrict__ skip,
                  const float* __restrict__ Wb, float* __restrict__ xnext,
                  int N, int do_relu)
{
    __shared__ float sWb[3 * DHID];
    for (int i = threadIdx.x; i < 3 * DHID; i += blockDim.x) sWb[i] = Wb[i];
    __syncthreads();

    const int n = blockIdx.x * blockDim.x + threadIdx.x;
    if (n >= N) return;
    float z = 0.0f;
    for (int c = 0; c < DHID; ++c) {
        const float o  = out [(size_t)n * DHID + c];
        const float xr = skip[(size_t)n * DHID + c];
        z += o * sWb[c] + xr * sWb[DHID + c] + (o - xr) * sWb[2 * DHID + c];
    }
    const float beta = 1.0f / (1.0f + expf(-z));
    for (int c = 0; c < DHID; ++c) {
        const float o  = out [(size_t)n * DHID + c];
        const float xr = skip[(size_t)n * DHID + c];
        float xn = beta * xr + (1.0f - beta) * o;
        if (do_relu) xn = fmaxf(xn, 0.0f);
        xnext[(size_t)n * DHID + c] = xn;
    }
}

// =====================================================================
// Final LayerNorm over 64 channels
// =====================================================================
__global__ __launch_bounds__(256)
void layer_norm(const float* __restrict__ x, const float* __restrict__ g,
                const float* __restrict__ b, float* __restrict__ y, int N)
{
    const int n = blockIdx.x * blockDim.x + threadIdx.x;
    if (n >= N) return;
    float mu = 0.0f;
    for (int c = 0; c < DHID; ++c) mu += x[(size_t)n * DHID + c];
    mu *= (1.0f / DHID);
    float var = 0.0f;
    for (int c = 0; c < DHID; ++c) {
        const float dd = x[(size_t)n * DHID + c] - mu;
        var += dd * dd;
    }
    var *= (1.0f / DHID);
    const float inv = 1.0f / sqrtf(var + 1e-5f);
    for (int c = 0; c < DHID; ++c)
        y[(size_t)n * DHID + c] = (x[(size_t)n * DHID + c] - mu) * inv * g[c] + b[c];
}

// =====================================================================
// Host launcher
// =====================================================================
extern "C" void kernel_launch(void* const* d_in, const int* in_sizes, int n_in,
                              void* d_out, int out_size, void* d_ws, size_t ws_size,
                              hipStream_t stream)
{
    const int N = N_NODES, E = N_EDGES;
    const size_t NM = (size_t)N * DHID;

    const float* xin   = (const float*)d_in[0];   // (N,7)
    const int*   eidx  = (const int*)  d_in[1];   // (2,E)
    const int*   src   = eidx;
    const int*   dstp  = eidx + E;
    const float* eattr = (const float*)d_in[2];   // (E,1)
    const float* cpn   = (const float*)d_in[3];   // (N,)
    const float* lng   = (const float*)d_in[37];
    const float* lnb   = (const float*)d_in[38];

    // workspace carve-up
    float* ws = (float*)d_ws;
    float* O      = ws;            ws += 4 * NM;  // fused [Q|K|V|S]
    float* outacc = ws;            ws += NM;
    float* xbuf0  = ws;            ws += NM;
    float* xbuf1  = ws;            ws += NM;
    float* xpad   = ws;            ws += (size_t)N * 8;
    float* Wcat   = ws;            ws += (size_t)DHID * 256;
    float* bcat   = ws;            ws += 256;
    float* alpha  = ws;            ws += (size_t)E * 4;
    float* denom  = ws;            ws += (size_t)N * 4;
    unsigned* segmax = (unsigned*)ws;

    float* Q  = O;
    float* Kb = O + NM;
    float* V  = O + 2 * NM;
    float* S  = O + 3 * NM;
    float* xbuf[2] = {xbuf0, xbuf1};

    const int blkE   = (E + 255) / 256;
    const int blkN   = (N + 255) / 256;
    const int blkM16 = N / 16;                    // N % 16 == 0

    pad_x<<<(N * 8 + 255) / 256, 256, 0, stream>>>(xin, xpad, N);

    const float* xcur = xpad;
    int K = 7, cur = 0;

    for (int l = 0; l < 3; ++l) {
        const int base = 4 + 11 * l;
        const float* Wq = (const float*)d_in[base + 0];
        const float* bq = (const float*)d_in[base + 1];
        const float* Wk = (const float*)d_in[base + 2];
        const float* bk = (const float*)d_in[base + 3];
        const float* Wv = (const float*)d_in[base + 4];
        const float* bv = (const float*)d_in[base + 5];
        const float* We = (const float*)d_in[base + 6];
        const float* Wsk= (const float*)d_in[base + 7];
        const float* bsk= (const float*)d_in[base + 8];
        const float* Wbt= (const float*)d_in[base + 9];
        const float* lam= (const float*)d_in[base + 10];

        const int H  = (l < 2) ? 4 : 1;
        const int KP = (l == 0) ? 8 : DHID;

        pack_weights<<<(KP * 256 + 255) / 256, 256, 0, stream>>>(
            Wq, Wk, Wv, Wsk, bq, bk, bv, bsk, Wcat, bcat, K, KP);

        if (l == 0)
            gemm_qkvs<8><<<blkM16, 256, 0, stream>>>(xcur, Wcat, bcat, O, N);
        else
            gemm_qkvs<DHID><<<blkM16, 256, 0, stream>>>(xcur, Wcat, bcat, O, N);

        seg_init<<<(int)((NM + 255) / 256), 256, 0, stream>>>(
            outacc, denom, segmax, (int)NM, N * H);

        if (l < 2) {
            edge_alpha<4,16><<<blkE, 256, 0, stream>>>(
                src, dstp, eattr, cpn, We, lam, Q, Kb, alpha, segmax, E);
            edge_exp<4><<<(E * 4 + 255) / 256, 256, 0, stream>>>(
                dstp, alpha, segmax, denom, E);
            edge_scatter<4,16><<<blkE, 256, 0, stream>>>(
                src, dstp, eattr, We, alpha, denom, V, outacc, E);
        } else {
            edge_alpha<1,64><<<blkE, 256, 0, stream>>>(
                src, dstp, eattr, cpn, We, lam, Q, Kb, alpha, segmax, E);
            edge_exp<1><<<blkE, 256, 0, stream>>>(
                dstp, alpha, segmax, denom, E);
            edge_scatter<1,64><<<blkE, 256, 0, stream>>>(
                src, dstp, eattr, We, alpha, denom, V, outacc, E);
        }

        node_combine<<<blkN, 256, 0, stream>>>(
            outacc, S, Wbt, xbuf[cur], N, (l < 2) ? 1 : 0);

        xcur = xbuf[cur];
        cur ^= 1;
        K = DHID;
    }

    layer_norm<<<blkN, 256, 0, stream>>>(xcur, lng, lnb, (float*)d_out, N);
}